// ConfusionMatrix_42554535969228
// MI455X (gfx1250) — compile-verified
//
#include <hip/hip_runtime.h>
#include <hip/hip_bf16.h>

typedef __attribute__((ext_vector_type(2))) float v2f;
typedef __attribute__((ext_vector_type(8))) float v8f;

#define NC 21
#define NB 8
#define NK (NB * NC)  // 168 stacked rows
#define HW (512 * 512)
#define EPSF 1e-8f
#define BLOCKS_PER_IMG 128
#define TPB 256
#define LDS_PITCH 22  // pad rows to avoid systematic bank patterns

// ---------------------------------------------------------------------------
// Kernel 0: zero the per-batch confusion-matrix accumulator in workspace.
// ---------------------------------------------------------------------------
__global__ void cm_zero_kernel(float* __restrict__ ws, int n) {
    int i = blockIdx.x * blockDim.x + threadIdx.x;
    if (i < n) ws[i] = 0.0f;
}

// ---------------------------------------------------------------------------
// Kernel 1: streaming argmax + scatter accumulation (bandwidth-bound pass).
// Each block owns a contiguous pixel chunk of one batch image. LDS holds the
// block-local 21x21 (pitch 22) confusion tile; ds_add_f32 atomics accumulate,
// then one global_atomic_add_f32 pass merges into d_ws cm[b][21][21].
// 352 MB read once -> ~15 us floor at 23.3 TB/s.
// ---------------------------------------------------------------------------
__global__ __launch_bounds__(TPB) void cm_accum_kernel(
    const float* __restrict__ inp, const float* __restrict__ tgt,
    float* __restrict__ cm) {
    __shared__ float lcm[NC * LDS_PITCH];

    const int tid = threadIdx.x;
    for (int i = tid; i < NC * LDS_PITCH; i += TPB) lcm[i] = 0.0f;
    __syncthreads();

    const int b     = blockIdx.x / BLOCKS_PER_IMG;
    const int chunk = blockIdx.x % BLOCKS_PER_IMG;
    const int ppb   = HW / BLOCKS_PER_IMG;  // 2048 pixels per block
    const int p0    = chunk * ppb;

    const float* __restrict__ inb = inp + (size_t)b * NC * HW;
    const float* __restrict__ tgb = tgt + (size_t)b * NC * HW;

    for (int p = p0 + tid; p < p0 + ppb; p += TPB) {
        // argmax over classes (first-max semantics, matching jnp.argmax);
        // prefetch the target planes so phase 2 hits warm lines.
        float best = inb[p];
        int   bj   = 0;
        __builtin_prefetch(tgb + p, 0, 3);  // global_prefetch_b8
#pragma unroll
        for (int c = 1; c < NC; ++c) {
            float v = inb[(size_t)c * HW + p];
            __builtin_prefetch(tgb + (size_t)c * HW + p, 0, 3);
            bool gt = v > best;
            best = gt ? v : best;
            bj   = gt ? c : bj;
        }
        // scatter target mass of every class into predicted column bj
#pragma unroll
        for (int c = 0; c < NC; ++c) {
            float t = tgb[(size_t)c * HW + p];
            atomicAdd(&lcm[c * LDS_PITCH + bj], t);  // ds_add_f32
        }
    }
    __syncthreads();

    float* __restrict__ dst = cm + (size_t)b * NC * NC;
    for (int i = tid; i < NC * NC; i += TPB) {
        atomicAdd(&dst[i], lcm[(i / NC) * LDS_PITCH + (i % NC)]);  // global_atomic_add_f32
    }
}

// ---------------------------------------------------------------------------
// Kernel 2: finalize as a WMMA f32 GEMM (single wave32), fully LDS-resident.
//   out[21x21] = W[21x168] x Y[168x21]
//   Y[(b,i'),j] = cm[b][i'][j]      (staged into LDS once, float4 copy)
//   W[i,(b,i')] = (i==i') * 0.125 / (rowsum[b][i] + eps)
// rowsum[b][i] == sum_j cm[b][i][j] exactly (one-hot columns partition pixels).
// Inner loop: ds_load operands + cndmask selects + 4x V_WMMA_F32_16X16X4_F32.
// No divergent loads -> EXEC stays all-1s around the WMMAs.
// ---------------------------------------------------------------------------
__device__ inline v8f wmma4(v2f a, v2f b, v8f c) {
    return __builtin_amdgcn_wmma_f32_16x16x4_f32(
        /*neg_a=*/false, a, /*neg_b=*/false, b,
        /*c_mod=*/(short)0, c, /*reuse_a=*/false, /*reuse_b=*/false);
}

__global__ __launch_bounds__(32) void cm_finalize_kernel(
    const float* __restrict__ cm, float* __restrict__ out) {
    __shared__ float4 lcmv[(NK * NC) / 4];  // 3528 floats, 16B-aligned
    __shared__ float  rs[NK];               // pre-scaled row normalizers
    __shared__ int    cls[NK];              // cls[k] = k % 21 (kills div/mod in loop)
    float* lcm = (float*)lcmv;

    const int lane = threadIdx.x;  // 0..31, wave32

    // Stage the whole cm into LDS (coalesced float4), build class table.
    const float4* __restrict__ src = (const float4*)cm;
    for (int i = lane; i < (NK * NC) / 4; i += 32) lcmv[i] = src[i];
    for (int k = lane; k < NK; k += 32) cls[k] = k - (k / NC) * NC;
    __syncthreads();

    // Row normalizers from LDS rows (fold 1/B mean in).
    for (int k = lane; k < NK; k += 32) {
        float s = 0.0f;
#pragma unroll
        for (int j = 0; j < NC; ++j) s += lcm[k * NC + j];
        rs[k] = 0.125f / (s + EPSF);
    }
    __syncthreads();

    const int  half = lane >> 4;  // K-pair selector within the 4-wide K step
    const int  m    = lane & 15;  // row (A) / col (B) within a 16-tile
    const int  n1   = m + 16;     // column in the ni=1 tile
    const bool has1 = n1 < NC;
    const int  n1c  = has1 ? n1 : (NC - 1);  // clamped (in-bounds) index

    v8f acc00 = {}, acc01 = {}, acc10 = {}, acc11 = {};

    for (int k0 = 0; k0 < NK; k0 += 4) {
        const int   kA = k0 + 2 * half;  // K index in VGPR0 of this lane
        const int   kB = kA + 1;         // K index in VGPR1
        const float sA = rs[kA];
        const float sB = rs[kB];
        const int   iA = cls[kA];
        const int   iB = cls[kB];

        // A operand (selector matrix): tiles mi=0 (rows 0..15), mi=1 (16..20)
        v2f a0, a1, b0, b1;
        a0.x = (iA == m) ? sA : 0.0f;
        a0.y = (iB == m) ? sB : 0.0f;
        a1.x = (iA == m + 16) ? sA : 0.0f;
        a1.y = (iB == m + 16) ? sB : 0.0f;

        // B operand: raw cm rows kA/kB from LDS; ni=1 columns zero-padded via
        // clamped index + select (no divergent loads).
        const float* rowA = lcm + kA * NC;
        const float* rowB = lcm + kB * NC;
        b0.x = rowA[m];
        b0.y = rowB[m];
        const float vA1 = rowA[n1c];
        const float vB1 = rowB[n1c];
        b1.x = has1 ? vA1 : 0.0f;
        b1.y = has1 ? vB1 : 0.0f;

        acc00 = wmma4(a0, b0, acc00);
        acc01 = wmma4(a0, b1, acc01);
        acc10 = wmma4(a1, b0, acc10);
        acc11 = wmma4(a1, b1, acc11);
    }

    // D layout: VGPR r holds row (8*half + r) of the 16x16 tile, col = m.
#pragma unroll
    for (int r = 0; r < 8; ++r) {
        const int row0 = half * 8 + r;       // tile mi=0
        const int row1 = 16 + half * 8 + r;  // tile mi=1
        if (row0 < NC) {
            out[row0 * NC + m] = acc00[r];
            if (has1) out[row0 * NC + n1] = acc01[r];
        }
        if (row1 < NC) {
            out[row1 * NC + m] = acc10[r];
            if (has1) out[row1 * NC + n1] = acc11[r];
        }
    }
}

// ---------------------------------------------------------------------------
extern "C" void kernel_launch(void* const* d_in, const int* in_sizes, int n_in,
                              void* d_out, int out_size, void* d_ws, size_t ws_size,
                              hipStream_t stream) {
    const float* inp = (const float*)d_in[0];  // [8,21,512,512] f32
    const float* tgt = (const float*)d_in[1];  // [8,21,512,512] f32
    float* out = (float*)d_out;                // [21,21] f32
    float* cm  = (float*)d_ws;                 // [8,21,21] f32 accumulator

    const int ncm = NB * NC * NC;  // 3528
    cm_zero_kernel<<<(ncm + 255) / 256, 256, 0, stream>>>(cm, ncm);
    cm_accum_kernel<<<NB * BLOCKS_PER_IMG, TPB, 0, stream>>>(inp, tgt, cm);
    cm_finalize_kernel<<<1, 32, 0, stream>>>(cm, out);
}